// MultiHeadedAttentionGaan_39805756900159
// MI455X (gfx1250) — compile-verified
//
#include <hip/hip_runtime.h>
#include <math.h>

// ---------------- problem constants ----------------
#define BB 8
#define LL 320
#define DD 512
#define HH 8
#define DKV 64
#define NDIM 64
#define NREL 50
#define NARC 4
#define RELD 32
#define NEGV -1e18f

typedef __attribute__((ext_vector_type(2))) float v2f;
typedef __attribute__((ext_vector_type(8))) float v8f;

// fp32 WMMA: D(16x16,f32) = A(16x4,f32) x B(4x16,f32) + C
__device__ __forceinline__ v8f wmma_f32x4(v2f a, v2f b, v8f c) {
  return __builtin_amdgcn_wmma_f32_16x16x4_f32(false, a, false, b, (short)0, c,
                                               false, false);
}

// ================= generic tiled GEMM: C[M,N] = act(X[M,K] @ W[K,N] + bias) =
// block = 128 threads (4 wave32), 64x64 output tile. Each wave owns a full
// 16-row x 64-col band (4 accumulators) so every A fragment feeds 4 WMMAs.
// K staged through LDS in chunks of 16; next chunk prefetched during compute.
__global__ void gemm_wmma_f32(const float* __restrict__ X,
                              const float* __restrict__ W,
                              const float* __restrict__ bias,
                              float* __restrict__ C,
                              int M, int N, int K, int act) {
  __shared__ float As[64 * 16];
  __shared__ float Bs[16 * 64];

  const int tid  = threadIdx.x;
  const int wave = tid >> 5;
  const int lane = tid & 31;
  const int half = lane >> 4;
  const int l16  = lane & 15;

  const int m0 = blockIdx.x * 64;
  const int n0 = blockIdx.y * 64;

  v8f acc[4] = {{}, {}, {}, {}};

  for (int kc = 0; kc < K; kc += 16) {
    // cooperative stage of A(64x16) and W(16x64)
    for (int i = tid; i < 1024; i += 128) {
      int r = i >> 4, c = i & 15;
      As[i] = X[(size_t)(m0 + r) * K + kc + c];
    }
    for (int i = tid; i < 1024; i += 128) {
      int r = i >> 6, c = i & 63;
      Bs[i] = W[(size_t)(kc + r) * N + n0 + c];
    }
    // prefetch next chunk (global_prefetch_b8) while WMMAs run
    if (kc + 16 < K) {
      __builtin_prefetch(&X[(size_t)(m0 + (tid >> 1)) * K + kc + 16], 0, 3);
      __builtin_prefetch(&W[(size_t)(kc + 16 + (tid >> 3)) * N + n0], 0, 3);
    }
    __syncthreads();

    #pragma unroll
    for (int kk = 0; kk < 4; ++kk) {
      const int kb = kk * 4 + 2 * half;
      v2f a;
      a.x = As[(wave * 16 + l16) * 16 + kb + 0];
      a.y = As[(wave * 16 + l16) * 16 + kb + 1];
      #pragma unroll
      for (int s = 0; s < 4; ++s) {
        v2f b;
        b.x = Bs[(kb + 0) * 64 + s * 16 + l16];
        b.y = Bs[(kb + 1) * 64 + s * 16 + l16];
        acc[s] = wmma_f32x4(a, b, acc[s]);
      }
    }
    __syncthreads();
  }

  #pragma unroll
  for (int s = 0; s < 4; ++s) {
    #pragma unroll
    for (int r = 0; r < 8; ++r) {
      const int row  = m0 + wave * 16 + r + 8 * half;
      const int colg = n0 + s * 16 + l16;
      float v = acc[s][r] + bias[colg];
      if (act == 1) v = v > 0.0f ? v : 0.1f * v;
      C[(size_t)row * N + colg] = v;
    }
  }
}

// ============ fused attention: one block = (b, h, 16 query rows) ============
// block = 128 threads (4 wave32). K/V staged in coalesced 64x64 LDS chunks,
// scores via WMMA, dep-rel/arc terms via precomputed per-row lookup tables,
// softmax + ds_add_f32 histogram in LDS, context via WMMA + micro-GEMM of
// histogram against value embeddings.
__global__ void attn_fused(const float* __restrict__ qbuf,
                           const float* __restrict__ kbuf,
                           const float* __restrict__ vbuf,
                           const int* __restrict__ mask,
                           const int* __restrict__ adj,
                           const int* __restrict__ deprel,
                           const int* __restrict__ deparc,
                           const float* __restrict__ Erel_k,
                           const float* __restrict__ Earc_k,
                           const float* __restrict__ Erel_v,
                           const float* __restrict__ Earc_v,
                           float* __restrict__ ctxbuf,
                           float* __restrict__ attn0) {
  __shared__ float qs[16 * 64];
  __shared__ float kv[64 * 64];     // staged K / V chunk (16 KB)
  __shared__ float sc[16 * LL];     // scores -> attn
  __shared__ float srel[16 * NREL];
  __shared__ float sarc[16 * NARC];
  __shared__ float wrel[16 * NREL];
  __shared__ float warc[16 * NARC];
  __shared__ float ctxadd[16 * 64];
  __shared__ float red[16 * 8];
  __shared__ float rowmax[16];
  __shared__ float rowinv[16];

  const int tid  = threadIdx.x;
  const int wave = tid >> 5;
  const int lane = tid & 31;
  const int half = lane >> 4;
  const int l16  = lane & 15;

  const int bid = blockIdx.x;
  const int qt  = bid % (LL / 16);
  const int h   = (bid / (LL / 16)) % HH;
  const int b   = bid / ((LL / 16) * HH);
  const int q0  = qt * 16;

  // ---- stage 16x64 query tile ----
  for (int i = tid; i < 16 * 64; i += 128) {
    qs[i] = qbuf[(size_t)(b * LL + q0 + (i >> 6)) * DD + h * 64 + (i & 63)];
  }
  __syncthreads();

  // ---- per-row relation/arc score tables: s_rel = q[:, :32] @ Erel_k^T ----
  for (int i = tid; i < 16 * NREL; i += 128) {
    int r = i / NREL, e = i % NREL;
    float s = 0.f;
    #pragma unroll 8
    for (int d = 0; d < RELD; ++d) s += qs[r * 64 + d] * Erel_k[e * RELD + d];
    srel[i] = s;
  }
  if (tid < 16 * NARC) {
    int r = tid / NARC, e = tid % NARC;
    float s = 0.f;
    #pragma unroll 8
    for (int d = 0; d < RELD; ++d)
      s += qs[r * 64 + 32 + d] * Earc_k[e * RELD + d];
    sarc[tid] = s;
  }
  __syncthreads();

  // ---- scores: q @ k^T, K staged in 64-key chunks (coalesced) ----
  for (int kc = 0; kc < LL; kc += 64) {
    for (int i = tid; i < 64 * 64; i += 128) {
      kv[i] = kbuf[(size_t)(b * LL + kc + (i >> 6)) * DD + h * 64 + (i & 63)];
    }
    __syncthreads();

    const int k0 = wave * 16;  // 16-key band within chunk, one per wave
    v8f acc = {};
    #pragma unroll 4
    for (int kk = 0; kk < 64; kk += 4) {
      const int kb = kk + 2 * half;
      v2f a;
      a.x = qs[l16 * 64 + kb + 0];
      a.y = qs[l16 * 64 + kb + 1];
      v2f bk;
      bk.x = kv[(k0 + l16) * 64 + kb + 0];
      bk.y = kv[(k0 + l16) * 64 + kb + 1];
      acc = wmma_f32x4(a, bk, acc);
    }
    #pragma unroll
    for (int r = 0; r < 8; ++r) {
      const int srow = r + 8 * half;
      const int col  = kc + k0 + l16;
      const size_t idx = (size_t)(b * LL + q0 + srow) * LL + col;
      const int rel = deprel[idx];
      const int arc = deparc[idx];
      float s = (acc[r] + srel[srow * NREL + rel] + sarc[srow * NARC + arc]) *
                0.125f;  // 1/sqrt(64)
      const bool att = (mask[idx] != 0) && (adj[idx] != 0);
      sc[srow * LL + col] = att ? s : NEGV;
    }
    __syncthreads();
  }

  // ---- softmax over 320 cols (8 threads per row, 40 cols each) ----
  const int r8  = tid >> 3;
  const int sub = tid & 7;
  const int c0  = sub * 40;
  float m = -3.0e38f;
  for (int c = c0; c < c0 + 40; ++c) m = fmaxf(m, sc[r8 * LL + c]);
  red[r8 * 8 + sub] = m;
  __syncthreads();
  if (sub == 0) {
    float mm = red[r8 * 8];
    #pragma unroll
    for (int j = 1; j < 8; ++j) mm = fmaxf(mm, red[r8 * 8 + j]);
    rowmax[r8] = mm;
  }
  // zero histograms while waiting
  for (int i = tid; i < 16 * NREL; i += 128) wrel[i] = 0.f;
  if (tid < 16 * NARC) warc[tid] = 0.f;
  __syncthreads();

  float sum = 0.f;
  {
    const float mm = rowmax[r8];
    for (int c = c0; c < c0 + 40; ++c) {
      float e = __expf(sc[r8 * LL + c] - mm);
      sc[r8 * LL + c] = e;
      sum += e;
    }
  }
  red[r8 * 8 + sub] = sum;
  __syncthreads();
  if (sub == 0) {
    float ss = 0.f;
    #pragma unroll
    for (int j = 0; j < 8; ++j) ss += red[r8 * 8 + j];
    rowinv[r8] = 1.0f / ss;
  }
  __syncthreads();

  // ---- normalize, histogram into rel/arc bins, emit head-0 attention ----
  {
    const float inv = rowinv[r8];
    for (int c = c0; c < c0 + 40; ++c) {
      float a = sc[r8 * LL + c] * inv;
      sc[r8 * LL + c] = a;
      const size_t idx = (size_t)(b * LL + q0 + r8) * LL + c;
      atomicAdd(&wrel[r8 * NREL + deprel[idx]], a);
      atomicAdd(&warc[r8 * NARC + deparc[idx]], a);
      if (h == 0) attn0[idx] = a;
    }
  }
  __syncthreads();

  // ---- edge-value term: w_rel @ Erel_v | w_arc @ Earc_v ----
  for (int i = tid; i < 16 * 64; i += 128) {
    const int row = i >> 6, d = i & 63;
    float s = 0.f;
    if (d < RELD) {
      for (int e = 0; e < NREL; ++e)
        s += wrel[row * NREL + e] * Erel_v[e * RELD + d];
    } else {
      const int dd = d - RELD;
      #pragma unroll
      for (int e = 0; e < NARC; ++e)
        s += warc[row * NARC + e] * Earc_v[e * RELD + dd];
    }
    ctxadd[i] = s;
  }
  __syncthreads();

  // ---- context: attn(16x320) @ v(320x64); V staged in 64-row chunks ----
  {
    const int n0 = wave * 16;  // 16 output cols per wave
    v8f acc = {};
    for (int kc = 0; kc < LL; kc += 64) {
      for (int i = tid; i < 64 * 64; i += 128) {
        kv[i] = vbuf[(size_t)(b * LL + kc + (i >> 6)) * DD + h * 64 + (i & 63)];
      }
      __syncthreads();

      #pragma unroll 4
      for (int kk = 0; kk < 64; kk += 4) {
        const int kb = kk + 2 * half;
        v2f a;
        a.x = sc[l16 * LL + kc + kb + 0];
        a.y = sc[l16 * LL + kc + kb + 1];
        v2f bv;
        bv.x = kv[(kb + 0) * 64 + n0 + l16];
        bv.y = kv[(kb + 1) * 64 + n0 + l16];
        acc = wmma_f32x4(a, bv, acc);
      }
      __syncthreads();
    }
    #pragma unroll
    for (int r = 0; r < 8; ++r) {
      const int row = r + 8 * half;
      ctxbuf[(size_t)(b * LL + q0 + row) * DD + h * 64 + n0 + l16] =
          acc[r] + ctxadd[row * 64 + n0 + l16];
    }
  }
}

// ======== GaAN gate: masked max-pool of neigh + sigmoid([q, mx] @ Wg) ========
__global__ void gaan_gate(const float* __restrict__ query,
                          const float* __restrict__ neigh,
                          const int* __restrict__ mask,
                          const int* __restrict__ adj,
                          const float* __restrict__ Wg,
                          const float* __restrict__ bg,
                          float* __restrict__ gates) {
  __shared__ float mx[NDIM];
  const int bq = blockIdx.x;   // b*L + q
  const int b  = bq / LL;
  const int q  = bq % LL;
  const int d  = threadIdx.x;  // 64 threads

  float m = NEGV;
  const size_t rowbase = (size_t)(b * LL + q) * LL;
  for (int k = 0; k < LL; ++k) {
    if (mask[rowbase + k] && adj[rowbase + k])
      m = fmaxf(m, neigh[(size_t)(b * LL + k) * NDIM + d]);
  }
  mx[d] = m;
  __syncthreads();

  if (d < HH) {
    float s = bg[d];
    const float* qrow = &query[(size_t)bq * DD];
    for (int j = 0; j < DD; ++j) s += qrow[j] * Wg[j * HH + d];
    #pragma unroll 8
    for (int j = 0; j < NDIM; ++j) s += mx[j] * Wg[(DD + j) * HH + d];
    gates[(size_t)bq * HH + d] = 1.0f / (1.0f + __expf(-s));
  }
}

// ======== apply gate to context ========
__global__ void gate_scale(float* __restrict__ ctx,
                           const float* __restrict__ gates) {
  const size_t i = (size_t)blockIdx.x * blockDim.x + threadIdx.x;
  if (i >= (size_t)BB * LL * DD) return;
  const size_t row = i / DD;
  const int hh = (int)((i % DD) / DKV);
  ctx[i] *= gates[row * HH + hh];
}

// =============================== launcher ===================================
extern "C" void kernel_launch(void* const* d_in, const int* in_sizes, int n_in,
                              void* d_out, int out_size, void* d_ws,
                              size_t ws_size, hipStream_t stream) {
  const float* key   = (const float*)d_in[0];
  const float* value = (const float*)d_in[1];
  const float* query = (const float*)d_in[2];
  const int* mask    = (const int*)d_in[3];
  const int* adj     = (const int*)d_in[4];
  const int* deprel  = (const int*)d_in[5];
  const int* deparc  = (const int*)d_in[6];
  const float* Wk = (const float*)d_in[14];
  const float* bk = (const float*)d_in[15];
  const float* Wq = (const float*)d_in[16];
  const float* bq = (const float*)d_in[17];
  const float* Wv = (const float*)d_in[18];
  const float* bv = (const float*)d_in[19];
  const float* Wn = (const float*)d_in[20];
  const float* bn = (const float*)d_in[21];
  const float* Wg = (const float*)d_in[22];
  const float* bg = (const float*)d_in[23];
  const float* Wo = (const float*)d_in[24];
  const float* bo = (const float*)d_in[25];
  const float* Erel_k = (const float*)d_in[26];
  const float* Earc_k = (const float*)d_in[27];
  const float* Erel_v = (const float*)d_in[28];
  const float* Earc_v = (const float*)d_in[29];

  float* out   = (float*)d_out;                      // (B,L,D)
  float* attn0 = out + (size_t)BB * LL * DD;         // (B,L,L)

  const size_t M = (size_t)BB * LL;                  // 2560
  float* ws    = (float*)d_ws;
  float* qb    = ws;
  float* kb    = qb + M * DD;
  float* vb    = kb + M * DD;
  float* ctx   = vb + M * DD;
  float* neigh = ctx + M * DD;
  float* gates = neigh + M * NDIM;

  dim3 gProj((unsigned)(M / 64), DD / 64);
  gemm_wmma_f32<<<gProj, 128, 0, stream>>>(key,   Wk, bk, kb, (int)M, DD, DD, 0);
  gemm_wmma_f32<<<gProj, 128, 0, stream>>>(query, Wq, bq, qb, (int)M, DD, DD, 0);
  gemm_wmma_f32<<<gProj, 128, 0, stream>>>(value, Wv, bv, vb, (int)M, DD, DD, 1);
  gemm_wmma_f32<<<dim3((unsigned)(M / 64), 1), 128, 0, stream>>>(
      value, Wn, bn, neigh, (int)M, NDIM, DD, 0);

  attn_fused<<<BB * HH * (LL / 16), 128, 0, stream>>>(
      qb, kb, vb, mask, adj, deprel, deparc, Erel_k, Earc_k, Erel_v, Earc_v,
      ctx, attn0);

  gaan_gate<<<(unsigned)M, 64, 0, stream>>>(query, neigh, mask, adj, Wg, bg,
                                            gates);
  gate_scale<<<(unsigned)((M * DD + 255) / 256), 256, 0, stream>>>(ctx, gates);

  gemm_wmma_f32<<<gProj, 128, 0, stream>>>(ctx, Wo, bo, out, (int)M, DD, DD, 0);
}